// PointTPS_17463337025574
// MI455X (gfx1250) — compile-verified
//
#include <hip/hip_runtime.h>

// TPS point warp for MI455X (gfx1250, wave32).
// z[p] = sum_k phi(|kp_k - p|^2) * W[k] + W[1024] + W[1025]*px + W[1026]*py
// phi(d2) = 0.5 * d2 * ln(d2) = (0.5*ln2) * d2 * log2(d2)
//
// Reduction over k mapped onto V_WMMA_F32_16X16X4_F32:
//   A (16x4 f32)  = phi tile: M = 16 query points, K = 4 control points
//   B (4x16 f32)  = W rows:   col 0 = Wx, col 1 = Wy, cols 2..15 = 0
//   C/D (16x16)   = f32 accumulator; only columns 0,1 meaningful
// Affine terms folded in as one extra (peeled) WMMA with K rows
// {1, px, py, 0} against W rows 1024..1026 (row 1027 zero-padded).
//
// B fragment is built branch-free: lane column n=m selects a loop-invariant
// base pointer (wx for col 0, wy for col 1, don't-care otherwise) and scales
// by a loop-invariant 0/1 mask -> one ds_load_b64 + one pk_mul per chunk,
// no EXEC divergence anywhere near the WMMAs (EXEC must be all-1s).

typedef __attribute__((ext_vector_type(2))) float v2f;
typedef __attribute__((ext_vector_type(8))) float v8f;

#define N_KPS 1024
#define N_PTS 120000
#define WAVES_PER_BLOCK 8
#define PTS_PER_WAVE 16

__global__ __launch_bounds__(256) void tps_warp_kernel(
    const float* __restrict__ pts,   // [N_PTS][2]
    const float* __restrict__ kps,   // [N_KPS][2]
    const float* __restrict__ W,     // [N_KPS+3][2]
    float* __restrict__ out)         // [N_PTS][2]
{
    __shared__ float lds_kp[2 * N_KPS];      // interleaved kx,ky   (8 KB)
    __shared__ float lds_wx[N_KPS + 4];      // Wx rows 0..1027 (1027 = 0)
    __shared__ float lds_wy[N_KPS + 4];      // Wy rows 0..1027 (1027 = 0)
    __shared__ float lds_stage[WAVES_PER_BLOCK * PTS_PER_WAVE * 2];

    const int tid  = threadIdx.x;
    const int lane = tid & 31;
    const int wave = tid >> 5;
    const int m    = lane & 15;   // query-point row (M) == output column (N)
    const int half = lane >> 4;   // 0 -> K rows {k,k+1}; 1 -> K rows {k+2,k+3}

    // ---- cooperative stage of kps and de-interleaved W into LDS ----
    for (int i = tid; i < 2 * N_KPS; i += 256) lds_kp[i] = kps[i];
    for (int i = tid; i < N_KPS + 3; i += 256) {
        lds_wx[i] = W[2 * i + 0];
        lds_wy[i] = W[2 * i + 1];
    }
    if (tid == 0) { lds_wx[N_KPS + 3] = 0.0f; lds_wy[N_KPS + 3] = 0.0f; }
    __syncthreads();

    // Whole 16-point tiles (N_PTS % 16 == 0) -> validity is wave-uniform.
    // Invalid tail waves redundantly compute tile 0 with full EXEC and skip
    // only the final store.
    const int tileBase = (blockIdx.x * WAVES_PER_BLOCK + wave) * PTS_PER_WAVE;
    const bool valid = (tileBase < N_PTS);
    const int tb = valid ? tileBase : 0;

    const float px = pts[(tb + m) * 2 + 0];
    const float py = pts[(tb + m) * 2 + 1];

    // Loop-invariant B-column setup: column 0 reads Wx, column 1 reads Wy,
    // all other columns read don't-care data and are zeroed by bmask.
    const float bmask = (m < 2) ? 1.0f : 0.0f;
    const float* wsel = (m == 1) ? lds_wy : lds_wx;

    v8f acc = {0.f, 0.f, 0.f, 0.f, 0.f, 0.f, 0.f, 0.f};
    const float C = 0.34657359027997264f;  // 0.5*ln(2): phi = C*d2*log2(d2)

    // ---- main K loop: 256 chunks of 4 control points, 1 WMMA each ----
    #pragma unroll 4
    for (int k0 = 0; k0 < N_KPS; k0 += 4) {
        const int kA = k0 + half * 2;     // first of this lane's two K rows

        // control points for K rows kA, kA+1 (4 consecutive floats -> b128)
        const float kx0 = lds_kp[2 * kA + 0];
        const float ky0 = lds_kp[2 * kA + 1];
        const float kx1 = lds_kp[2 * kA + 2];
        const float ky1 = lds_kp[2 * kA + 3];

        const float dx0 = kx0 - px, dy0 = ky0 - py;
        const float dx1 = kx1 - px, dy1 = ky1 - py;
        float d20 = dx0 * dx0 + dy0 * dy0;
        float d21 = dx1 * dx1 + dy1 * dy1;
        d20 = (d20 == 0.0f) ? 1.0f : d20;
        d21 = (d21 == 0.0f) ? 1.0f : d21;

        v2f a;                                   // A fragment: phi(m, kA..kA+1)
        a.x = C * d20 * __log2f(d20);            // v_log_f32
        a.y = C * d21 * __log2f(d21);

        // B fragment: rows kA, kA+1 at column n=m (adjacent pair -> b64)
        v2f b;
        b.x = bmask * wsel[kA];
        b.y = bmask * wsel[kA + 1];

        acc = __builtin_amdgcn_wmma_f32_16x16x4_f32(
            /*neg_a=*/false, a, /*neg_b=*/false, b,
            /*c_mod=*/(short)0, acc, /*reuse_a=*/false, /*reuse_b=*/false);
    }

    // ---- peeled affine chunk: virtual K rows 1024..1027 ----
    {
        const int kA = N_KPS + half * 2;
        // A rows: {1, px} (rows 1024,1025) / {py, 0} (rows 1026,1027)
        v2f a;
        a.x = (half == 0) ? 1.0f : py;
        a.y = (half == 0) ? px   : 0.0f;
        v2f b;
        b.x = bmask * wsel[kA];
        b.y = bmask * wsel[kA + 1];
        acc = __builtin_amdgcn_wmma_f32_16x16x4_f32(
            false, a, false, b, (short)0, acc, false, false);
    }

    // ---- D -> output: lane holds rows r + 8*half (r=0..7) at column n=m.
    // Columns 0 (zx) and 1 (zy) are the answer; stage through LDS.
    if (m < 2) {
        #pragma unroll
        for (int r = 0; r < 8; ++r) {
            const int row = r + half * 8;
            lds_stage[wave * 32 + row * 2 + m] = acc[r];
        }
    }
    __syncthreads();   // all waves reach this (divergence closed above)

    if (valid && lane < 16) {
        const float zx = lds_stage[wave * 32 + m * 2 + 0];
        const float zy = lds_stage[wave * 32 + m * 2 + 1];
        out[(tileBase + m) * 2 + 0] = px + zx;
        out[(tileBase + m) * 2 + 1] = py + zy;
    }
}

extern "C" void kernel_launch(void* const* d_in, const int* in_sizes, int n_in,
                              void* d_out, int out_size, void* d_ws, size_t ws_size,
                              hipStream_t stream) {
    (void)in_sizes; (void)n_in; (void)out_size; (void)d_ws; (void)ws_size;
    const float* pts = (const float*)d_in[0];   // [120000, 2]
    const float* kps = (const float*)d_in[1];   // [1024, 2]
    const float* W   = (const float*)d_in[2];   // [1027, 2]
    float* out = (float*)d_out;                 // [120000, 2]

    const int tiles  = N_PTS / PTS_PER_WAVE;                             // 7500
    const int blocks = (tiles + WAVES_PER_BLOCK - 1) / WAVES_PER_BLOCK;  // 938
    tps_warp_kernel<<<blocks, 256, 0, stream>>>(pts, kps, W, out);
}